// MaxPoolMatMulStack_30537217475130
// MI455X (gfx1250) — compile-verified
//
#include <hip/hip_runtime.h>
#include <hip/hip_bf16.h>

#define N_ 4
#define C_ 19
#define H_ 512
#define W_ 512
#define TILE 64
#define LSTRIDE 72   /* floats per LDS row: body starts at col 4 -> 16B aligned */
#define LROWS 66
#define NUM_STEPS 256
#define EPS_ 1e-12f

// ---- CDNA5 async global->LDS data movers (gfx1250) ------------------------
__device__ __forceinline__ void async_ld_b128(unsigned lds_off, const float* g) {
  asm volatile("global_load_async_to_lds_b128 %0, %1, off"
               :: "v"(lds_off), "v"(g) : "memory");
}
__device__ __forceinline__ void async_ld_b32(unsigned lds_off, const float* g) {
  asm volatile("global_load_async_to_lds_b32 %0, %1, off"
               :: "v"(lds_off), "v"(g) : "memory");
}
__device__ __forceinline__ void wait_async0() {
  asm volatile("s_wait_asynccnt 0x0" ::: "memory");
}

// ---- cur0 = -x, borders = 1 (float4 path) ---------------------------------
__global__ __launch_bounds__(256) void init_kernel(const float4* __restrict__ x4,
                                                   float4* __restrict__ cur4) {
  const int idx = blockIdx.x * 256 + threadIdx.x;   // over total/4
  const int i = idx << 2;
  const int w = i & (W_ - 1);                       // 0,4,...,508
  const int h = (i >> 9) & (H_ - 1);
  const float4 xv = x4[idx];
  float4 v;
  v.x = -xv.x; v.y = -xv.y; v.z = -xv.z; v.w = -xv.w;
  if (h == 0 || h == H_ - 1) {
    v.x = 1.0f; v.y = 1.0f; v.z = 1.0f; v.w = 1.0f;
  } else {
    if (w == 0)       v.x = 1.0f;
    if (w == W_ - 4)  v.w = 1.0f;
  }
  cur4[idx] = v;
}

// ---- one step: dst = maxpool3x3_neginfpad(src) * x ------------------------
__global__ __launch_bounds__(256) void step_kernel(const float* __restrict__ src,
                                                   const float* __restrict__ x,
                                                   float* __restrict__ dst) {
  __shared__ float tile[LROWS * LSTRIDE];
  const int t   = threadIdx.x;
  const int tx0 = blockIdx.x * TILE;
  const int ty0 = blockIdx.y * TILE;
  const size_t pbase = (size_t)blockIdx.z * (size_t)(H_ * W_);
  const float* sp = src + pbase;
  const unsigned lds_base = (unsigned)(unsigned long long)(void*)tile;
  const float NEGINF = -__builtin_inff();

  // Stage 66x66 halo tile into LDS with async b128 (body) / b32 (halo cols).
  for (int job = t; job < LROWS * 16; job += 256) {
    const int lr = job >> 4;          // LDS row 0..65
    const int q  = job & 15;          // quad 0..15 (cols tx0+4q .. +3)
    const int gy = ty0 + lr - 1;
    const unsigned loff = lds_base + (unsigned)((lr * LSTRIDE + 4 + q * 4) * 4);
    if ((unsigned)gy < (unsigned)H_) {
      async_ld_b128(loff, sp + (size_t)gy * W_ + (size_t)(tx0 + q * 4));
    } else {
      float* p = &tile[lr * LSTRIDE + 4 + q * 4];
      p[0] = NEGINF; p[1] = NEGINF; p[2] = NEGINF; p[3] = NEGINF;
    }
  }
  if (t < 2 * LROWS) {
    const int lr   = t >> 1;
    const int side = t & 1;
    const int gy = ty0 + lr - 1;
    const int gx = side ? (tx0 + TILE) : (tx0 - 1);
    const int lc = side ? (4 + TILE) : 3;
    const unsigned loff = lds_base + (unsigned)((lr * LSTRIDE + lc) * 4);
    if ((unsigned)gy < (unsigned)H_ && (unsigned)gx < (unsigned)W_) {
      async_ld_b32(loff, sp + (size_t)gy * W_ + (size_t)gx);
    } else {
      tile[lr * LSTRIDE + lc] = NEGINF;
    }
  }
  wait_async0();        // this wave's async transfers landed in LDS
  __syncthreads();      // all waves' transfers visible

  // 4x4 micro-tile per thread: b128 LDS reads, b128 x loads, b128 stores.
  const int c0 = (t & 15) << 2;     // tile-local first output col (0..60)
  const int r0 = (t >> 4) << 2;     // tile-local first output row (0..60)

  auto hmax4 = [&](int lr) -> float4 {
    const float* row = &tile[lr * LSTRIDE + 4 + c0];
    const float4 v = *(const float4*)row;   // ds_load_b128 (16B aligned)
    const float lft = row[-1];
    const float rgt = row[4];
    float4 r;
    r.x = fmaxf(fmaxf(lft, v.x), v.y);
    r.y = fmaxf(fmaxf(v.x, v.y), v.z);
    r.z = fmaxf(fmaxf(v.y, v.z), v.w);
    r.w = fmaxf(fmaxf(v.z, v.w), rgt);
    return r;
  };

  const float* xp = x + pbase;
  float* dp = dst + pbase;
  size_t o = (size_t)(ty0 + r0) * W_ + (size_t)(tx0 + c0);
  __builtin_prefetch(xp + o, 0, 0);

  float4 ha = hmax4(r0);
  float4 hb = hmax4(r0 + 1);
#pragma unroll
  for (int i = 0; i < 4; ++i) {
    const float4 hc = hmax4(r0 + i + 2);
    const float4 xv = *(const float4*)(xp + o);
    float4 m;
    m.x = fmaxf(fmaxf(ha.x, hb.x), hc.x) * xv.x;
    m.y = fmaxf(fmaxf(ha.y, hb.y), hc.y) * xv.y;
    m.z = fmaxf(fmaxf(ha.z, hb.z), hc.z) * xv.z;
    m.w = fmaxf(fmaxf(ha.w, hb.w), hc.w) * xv.w;
    *(float4*)(dp + o) = m;
    ha = hb; hb = hc;
    o += W_;
  }
}

// ---- out = (cur / max(||cur||_2_overC, eps)) * x (float4 path) ------------
__global__ __launch_bounds__(256) void finalize_kernel(const float4* __restrict__ cur4,
                                                       const float4* __restrict__ x4,
                                                       float4* __restrict__ out4) {
  const int HW4 = (H_ * W_) / 4;                    // 65536
  const int pos = blockIdx.x * 256 + threadIdx.x;   // over N*HW4 = 262144
  const int b  = pos >> 16;
  const int hw = pos & (HW4 - 1);
  const size_t base = (size_t)b * C_ * HW4 + (size_t)hw;

  float4 v[C_];
  float4 ss; ss.x = 0.f; ss.y = 0.f; ss.z = 0.f; ss.w = 0.f;
#pragma unroll
  for (int c = 0; c < C_; ++c) {
    v[c] = cur4[base + (size_t)c * HW4];
    ss.x += v[c].x * v[c].x;
    ss.y += v[c].y * v[c].y;
    ss.z += v[c].z * v[c].z;
    ss.w += v[c].w * v[c].w;
  }
  float4 inv;
  inv.x = 1.0f / fmaxf(__builtin_sqrtf(ss.x), EPS_);
  inv.y = 1.0f / fmaxf(__builtin_sqrtf(ss.y), EPS_);
  inv.z = 1.0f / fmaxf(__builtin_sqrtf(ss.z), EPS_);
  inv.w = 1.0f / fmaxf(__builtin_sqrtf(ss.w), EPS_);
#pragma unroll
  for (int c = 0; c < C_; ++c) {
    const size_t oidx = base + (size_t)c * HW4;
    const float4 xv = x4[oidx];
    float4 r;
    r.x = v[c].x * inv.x * xv.x;
    r.y = v[c].y * inv.y * xv.y;
    r.z = v[c].z * inv.z * xv.z;
    r.w = v[c].w * inv.w * xv.w;
    out4[oidx] = r;
  }
}

extern "C" void kernel_launch(void* const* d_in, const int* in_sizes, int n_in,
                              void* d_out, int out_size, void* d_ws, size_t ws_size,
                              hipStream_t stream) {
  (void)in_sizes; (void)n_in; (void)out_size; (void)ws_size;
  const float* x = (const float*)d_in[0];
  float* out  = (float*)d_out;
  float* bufA = (float*)d_ws;   // 4*19*512*512*4 = ~76 MB scratch
  float* bufB = out;            // reuse d_out as the second ping-pong buffer

  const int total = N_ * C_ * H_ * W_;            // 19,922,944
  init_kernel<<<(total / 4) / 256, 256, 0, stream>>>((const float4*)x,
                                                     (float4*)bufA);

  dim3 grid(W_ / TILE, H_ / TILE, N_ * C_);       // (8, 8, 76)
  for (int i = 0; i < NUM_STEPS; ++i) {
    const float* s = (i & 1) ? bufB : bufA;
    float*       d = (i & 1) ? bufA : bufB;
    step_kernel<<<grid, 256, 0, stream>>>(s, x, d);
  }
  // NUM_STEPS even -> final cur lives in bufA; finalize fully rewrites d_out.
  finalize_kernel<<<(N_ * H_ * W_ / 4) / 256, 256, 0, stream>>>(
      (const float4*)bufA, (const float4*)x, (float4*)out);
}